// BareboneGIN_30786325577782
// MI455X (gfx1250) — compile-verified
//
#include <hip/hip_runtime.h>

#define HID     128
#define PITCH2  144          // LDS pair-row pitch in float2 (288 floats -> 32-bank row offset)
#define NLAYERS 3

typedef __attribute__((ext_vector_type(2))) float v2f;
typedef __attribute__((ext_vector_type(8))) float v8f;

// ---------------------------------------------------------------------------
// Copy X -> dst (float4 grid-stride).  Initializes agg with the GIN self term.
// ---------------------------------------------------------------------------
__global__ __launch_bounds__(256)
void gin_copy_f4(const float4* __restrict__ src, float4* __restrict__ dst, long long n4) {
    long long i = (long long)blockIdx.x * blockDim.x + threadIdx.x;
    long long stride = (long long)gridDim.x * blockDim.x;
    for (; i < n4; i += stride) dst[i] = src[i];
}

// ---------------------------------------------------------------------------
// Edge scatter-add: agg[dst[e]] += X[src[e]].  One wave (32 lanes) per edge,
// 4 floats per lane, hardware fp32 atomics (no return -> STOREcnt path).
// ---------------------------------------------------------------------------
__global__ __launch_bounds__(256)
void gin_scatter(const float* __restrict__ X,
                 const long long* __restrict__ srcIdx,
                 const long long* __restrict__ dstIdx,
                 float* __restrict__ agg, int nEdges) {
    long long t = (long long)blockIdx.x * blockDim.x + threadIdx.x;
    int e = (int)(t >> 5);
    if (e >= nEdges) return;
    int c = ((int)t & 31) * 4;
    const long long s = srcIdx[e];
    const long long d = dstIdx[e];
    const float4 v = *(const float4*)(X + s * HID + c);
    float* p = agg + d * HID + c;
    unsafeAtomicAdd(p + 0, v.x);
    unsafeAtomicAdd(p + 1, v.y);
    unsafeAtomicAdd(p + 2, v.z);
    unsafeAtomicAdd(p + 3, v.w);
}

// ---------------------------------------------------------------------------
// WMMA GEMM:  OUT[nRows,128] = act( Z[nRows,128] @ W[128,128] + b )
// Block = 128 threads = 4 waves; each wave -> 16 rows x 128 cols.
// V_WMMA_F32_16X16X4_F32, 32 K-steps.  W staged in LDS in K-PAIRED layout:
//   lW[(k/2)*PITCH2 + n] = float2{ W[k][n], W[k+1][n] }
// so each B fragment (K rows {kk,kk+1}, kk even) is ONE ds_load_b64 that
// lands directly in an even-aligned VGPR pair (no shuffle movs before wmma).
// A frag: lanes 0-15 hold M=lane, K={k,k+1}; lanes 16-31 hold M=lane-16, K={k+2,k+3}.
// C/D   : VGPR r -> M = rowBase + r + 8*(lane>=16), N = 16j + (lane&15).
// ---------------------------------------------------------------------------
__global__ __launch_bounds__(128)
void gin_mlp_gemm(const float* __restrict__ Z, const float* __restrict__ W,
                  const float* __restrict__ bias, float* __restrict__ OUT,
                  int nRows, int doRelu) {
    __shared__ float2 lW[(HID / 2) * PITCH2];
    __shared__ float  lB[HID];

    const int tid = threadIdx.x;
    // Stage W (row-major [k][n]) into LDS K-paired layout, float4 per iteration.
    for (int i = tid; i < (HID * HID) / 4; i += 128) {
        const float4 w = reinterpret_cast<const float4*>(W)[i];
        const int k = i >> 5;          // (i*4)/128
        const int n = (i & 31) << 2;   // (i*4)%128
        float* p = (float*)lW + (((k >> 1) * PITCH2 + n) << 1) + (k & 1);
        p[0] = w.x; p[2] = w.y; p[4] = w.z; p[6] = w.w;
    }
    if (tid < HID) lB[tid] = bias[tid];
    __syncthreads();

    const int wave    = tid >> 5;
    const int lane    = tid & 31;
    const int l16     = lane & 15;
    const int hiHalf  = lane >> 4;
    const int rowBase = blockIdx.x * 64 + wave * 16;

    int aRow = rowBase + l16;
    if (aRow >= nRows) aRow = nRows - 1;   // clamp; stores are guarded
    const float* aPtr = Z + (size_t)aRow * HID + hiHalf * 2;

    v8f acc[8];
#pragma unroll
    for (int j = 0; j < 8; ++j) acc[j] = (v8f){0.f,0.f,0.f,0.f,0.f,0.f,0.f,0.f};

    for (int k = 0; k < HID; k += 4) {
        const v2f a = *(const v2f*)(aPtr + k);
        const int kk = k + hiHalf * 2;              // even
        const v2f* bRow = (const v2f*)&lW[(kk >> 1) * PITCH2 + l16];
#pragma unroll
        for (int j = 0; j < 8; ++j) {
            const v2f b = bRow[j * 16];             // single ds_load_b64
            acc[j] = __builtin_amdgcn_wmma_f32_16x16x4_f32(
                false, a, false, b, (short)0, acc[j], false, false);
        }
    }

#pragma unroll
    for (int j = 0; j < 8; ++j) {
        const int n  = j * 16 + l16;
        const float bb = lB[n];
#pragma unroll
        for (int r = 0; r < 8; ++r) {
            const int row = rowBase + r + hiHalf * 8;
            float v = acc[j][r] + bb;
            if (doRelu) v = fmaxf(v, 0.f);
            if (row < nRows) OUT[(size_t)row * HID + n] = v;
        }
    }
}

// ---------------------------------------------------------------------------
// Per-graph sum pooling into pooled[g][layerOff + c]  (pooled pitch = 3*H)
// ---------------------------------------------------------------------------
__global__ __launch_bounds__(256)
void gin_pool(const float* __restrict__ X, const long long* __restrict__ batch,
              float* __restrict__ pooled, int nNodes, int layerOff) {
    long long t = (long long)blockIdx.x * blockDim.x + threadIdx.x;
    int node = (int)(t >> 7);
    if (node >= nNodes) return;
    int c = (int)t & (HID - 1);
    const long long g = batch[node];
    unsafeAtomicAdd(pooled + g * (HID * NLAYERS) + layerOff + c,
                    X[(size_t)node * HID + c]);
}

// ---------------------------------------------------------------------------
// Small dense layer: out[g][j] = act( in[g][:] @ W[:, j] + b[j] )
// ---------------------------------------------------------------------------
__global__ __launch_bounds__(256)
void cls_dense(const float* __restrict__ in, const float* __restrict__ W,
               const float* __restrict__ b, float* __restrict__ out,
               int G, int Kdim, int Ndim, int doRelu) {
    int t = blockIdx.x * blockDim.x + threadIdx.x;
    if (t >= G * Ndim) return;
    int g = t / Ndim, j = t - g * Ndim;
    float s = b[j];
    const float* row = in + (size_t)g * Kdim;
    for (int k = 0; k < Kdim; ++k) s += row[k] * W[(size_t)k * Ndim + j];
    out[t] = doRelu ? fmaxf(s, 0.f) : s;
}

// ---------------------------------------------------------------------------
extern "C" void kernel_launch(void* const* d_in, const int* in_sizes, int n_in,
                              void* d_out, int out_size, void* d_ws, size_t ws_size,
                              hipStream_t stream) {
    const float*     X     = (const float*)d_in[0];
    const long long* ei    = (const long long*)d_in[1];
    const long long* batch = (const long long*)d_in[2];
    const float*     w1    = (const float*)d_in[3];   // [3,128,128]
    const float*     b1    = (const float*)d_in[4];   // [3,128]
    const float*     w2    = (const float*)d_in[5];
    const float*     b2    = (const float*)d_in[6];
    const float*     cw1   = (const float*)d_in[7];   // [384,128]
    const float*     cb1   = (const float*)d_in[8];
    const float*     cw2   = (const float*)d_in[9];   // [128,128]
    const float*     cb2   = (const float*)d_in[10];
    const float*     cw3   = (const float*)d_in[11];  // [128,1]
    const float*     cb3   = (const float*)d_in[12];

    const int nNodes = in_sizes[0] / HID;
    const int nEdges = in_sizes[1] / 2;
    const int G      = out_size;                      // 512 graphs, 1 logit each

    const long long* srcIdx = ei;
    const long long* dstIdx = ei + nEdges;

    const size_t NH = (size_t)nNodes * HID;
    float* A      = (float*)d_ws;                     // layer output X
    float* Bz     = A + NH;                           // z = X + agg
    float* Ch     = Bz + NH;                          // MLP hidden
    float* pooled = Ch + NH;                          // [G, 3*128]
    float* h1     = pooled + (size_t)G * HID * NLAYERS;
    float* h2     = h1 + (size_t)G * HID;

    hipMemsetAsync(pooled, 0, (size_t)G * HID * NLAYERS * sizeof(float), stream);

    const int copyBlocks    = 2048;
    const int scatterBlocks = (int)(((long long)nEdges * 32 + 255) / 256);
    const int gemmBlocks    = (nNodes + 63) / 64;
    const int poolBlocks    = (int)(((long long)nNodes * HID + 255) / 256);

    const float* Xcur = X;
    for (int l = 0; l < NLAYERS; ++l) {
        gin_copy_f4<<<copyBlocks, 256, 0, stream>>>(
            (const float4*)Xcur, (float4*)Bz, (long long)(NH / 4));
        gin_scatter<<<scatterBlocks, 256, 0, stream>>>(Xcur, srcIdx, dstIdx, Bz, nEdges);
        gin_mlp_gemm<<<gemmBlocks, 128, 0, stream>>>(
            Bz, w1 + (size_t)l * HID * HID, b1 + (size_t)l * HID, Ch, nNodes, 1);
        gin_mlp_gemm<<<gemmBlocks, 128, 0, stream>>>(
            Ch, w2 + (size_t)l * HID * HID, b2 + (size_t)l * HID, A, nNodes, 1);
        gin_pool<<<poolBlocks, 256, 0, stream>>>(A, batch, pooled, nNodes, l * HID);
        Xcur = A;
    }

    cls_dense<<<(G * HID + 255) / 256, 256, 0, stream>>>(pooled, cw1, cb1, h1, G, HID * NLAYERS, HID, 1);
    cls_dense<<<(G * HID + 255) / 256, 256, 0, stream>>>(h1, cw2, cb2, h2, G, HID, HID, 1);
    cls_dense<<<(G + 255) / 256, 256, 0, stream>>>(h2, cw3, cb3, (float*)d_out, G, HID, 1, 0);
}